// KPlanesEncoder_89215060672780
// MI455X (gfx1250) — compile-verified
//
#include <hip/hip_runtime.h>

#define N_PTS   500000
#define FDIM    16
#define NPLANES 9

// 8-byte vector with 4-byte alignment: x0/x0+1 are adjacent in a row but the
// base has arbitrary dword parity. gfx1250 supports dword-aligned b64 loads.
typedef float f32x2 __attribute__((ext_vector_type(2), aligned(4)));

struct PlanePtrs { const float* p[NPLANES]; };

__global__ __launch_bounds__(256) void KPlanesEncoder_kernel(
    const float* __restrict__ x, PlanePtrs planes, float* __restrict__ out)
{
    const int n = blockIdx.x * 256 + threadIdx.x;
    if (n >= N_PTS) return;

    const int p  = blockIdx.y;            // plane index 0..8 (uniform)
    const int i  = p / 3;                 // first coord axis
    const int j  = p % 3;                 // resolution index: res = 128<<j
    const int W  = 128 << j;              // H == W == res
    const int i2 = (i == 2) ? 0 : i + 1;  // second coord axis

    const float rawx = x[3 * n + i];
    const float rawy = x[3 * n + i2];

    // Reference applies the align-corners transform TWICE:
    //   g  = (raw + 1) * (res-1)/2
    //   ix = (g   + 1) * 0.5 * (res-1)
    const float s  = 0.5f * (float)(W - 1);
    float ix = ((rawx + 1.0f) * s + 1.0f) * s;
    float iy = ((rawy + 1.0f) * s + 1.0f) * s;
    const float wmax = (float)(W - 1);
    ix = fminf(fmaxf(ix, 0.0f), wmax);
    iy = fminf(fmaxf(iy, 0.0f), wmax);

    // floor == trunc since ix,iy >= 0; clamp base to res-2 and recompute the
    // fractional weight (identical result to x1=min(x0+1,W-1) border rule).
    int x0 = (int)ix; if (x0 > W - 2) x0 = W - 2;
    int y0 = (int)iy; if (y0 > W - 2) y0 = W - 2;
    const float wx = ix - (float)x0;
    const float wy = iy - (float)y0;
    const float w00 = (1.0f - wx) * (1.0f - wy);
    const float w01 = wx * (1.0f - wy);
    const float w10 = (1.0f - wx) * wy;
    const float w11 = wx * wy;

    const float* base = planes.p[p] + ((size_t)y0 * (size_t)W + (size_t)x0);
    const size_t slab = (size_t)W * (size_t)W;   // per-feature stride

    float acc[FDIM];
#pragma unroll
    for (int f = 0; f < FDIM; ++f) {
        // top row (y0): texels x0, x0+1 ; bottom row (y0+1) via +W floats
        const f32x2 a = *(const f32x2*)(base + (size_t)f * slab);
        const f32x2 b = *(const f32x2*)(base + (size_t)f * slab + (size_t)W);
        acc[f] = fmaf(a.x, w00, fmaf(a.y, w01, fmaf(b.x, w10, b.y * w11)));
    }

    // out[f, p*N + n], row-major (16, 9N). Streaming, never re-read:
    // non-temporal stores keep the 66 MB plane set resident in the 192 MB L2.
    float* o = out + (size_t)p * (size_t)N_PTS + (size_t)n;
    const size_t frow = (size_t)NPLANES * (size_t)N_PTS;
#pragma unroll
    for (int f = 0; f < FDIM; ++f) {
        __builtin_nontemporal_store(acc[f], o + (size_t)f * frow);
    }
}

extern "C" void kernel_launch(void* const* d_in, const int* in_sizes, int n_in,
                              void* d_out, int out_size, void* d_ws, size_t ws_size,
                              hipStream_t stream) {
    const float* xpts = (const float*)d_in[0];
    PlanePtrs pl;
#pragma unroll
    for (int k = 0; k < NPLANES; ++k) pl.p[k] = (const float*)d_in[1 + k];

    dim3 block(256, 1, 1);
    dim3 grid((N_PTS + 255) / 256, NPLANES, 1);
    KPlanesEncoder_kernel<<<grid, block, 0, stream>>>(xpts, pl, (float*)d_out);
}